// _fasterRCNN_inc_bbox_distil_20873541059197
// MI455X (gfx1250) — compile-verified
//
#include <hip/hip_runtime.h>
#include <hip/hip_bf16.h>
#include <math.h>

// Problem constants (from reference setup_inputs)
#define N_ROIS   2000
#define N_CLS    21
#define N_JCLS   20      // classes 1..20 processed
#define N_GT     20
#define FEAT_C   512
#define FEAT_P   1900    // 38*50
#define NMS_THR   0.3f
#define SCORE_THR 0.5f
#define DEDUP_THR 0.3f

typedef __attribute__((ext_vector_type(2))) float v2f;
typedef __attribute__((ext_vector_type(8))) float v8f;

__device__ __forceinline__ float iou_f(float ax1, float ay1, float ax2, float ay2,
                                       float bx1, float by1, float bx2, float by2) {
    float areaA = (ax2 - ax1) * (ay2 - ay1);
    float areaB = (bx2 - bx1) * (by2 - by1);
    float lx = fmaxf(ax1, bx1), ly = fmaxf(ay1, by1);
    float rx = fminf(ax2, bx2), ry = fminf(ay2, by2);
    float w = fmaxf(rx - lx, 0.0f), h = fmaxf(ry - ly, 0.0f);
    float inter = w * h;
    return inter / (areaA + areaB - inter + 1e-6f);
}

// ---------------------------------------------------------------------------
// Kernel 1: bbox decode. pred stored as [C][N][4] in workspace.
// ---------------------------------------------------------------------------
__global__ void decode_kernel(const float* __restrict__ rois,
                              const float* __restrict__ bbox_pred,
                              const float* __restrict__ im_info,
                              float* __restrict__ pred) {
    int id = blockIdx.x * blockDim.x + threadIdx.x;
    if (id >= N_ROIS * N_CLS) return;
    int n = id / N_CLS, c = id % N_CLS;

    float x1 = rois[n * 5 + 1], y1 = rois[n * 5 + 2];
    float x2 = rois[n * 5 + 3], y2 = rois[n * 5 + 4];
    float w  = x2 - x1 + 1.0f, h = y2 - y1 + 1.0f;
    float cx = x1 + 0.5f * w,  cy = y1 + 0.5f * h;

    const float* d = bbox_pred + (n * N_CLS + c) * 4;
    float dx = d[0] * 0.1f, dy = d[1] * 0.1f;
    float dw = d[2] * 0.2f, dh = d[3] * 0.2f;

    float pcx = dx * w + cx, pcy = dy * h + cy;
    float pw = expf(dw) * w, ph = expf(dh) * h;

    float H = im_info[0], W = im_info[1];
    float ox1 = fminf(fmaxf(pcx - 0.5f * pw, 0.0f), W - 1.0f);
    float oy1 = fminf(fmaxf(pcy - 0.5f * ph, 0.0f), H - 1.0f);
    float ox2 = fminf(fmaxf(pcx + 0.5f * pw, 0.0f), W - 1.0f);
    float oy2 = fminf(fmaxf(pcy + 0.5f * ph, 0.0f), H - 1.0f);

    float* o = pred + ((size_t)c * N_ROIS + n) * 4;
    o[0] = ox1; o[1] = oy1; o[2] = ox2; o[3] = oy2;
}

// ---------------------------------------------------------------------------
// Kernel 2: per-class stable sort (LDS bitonic over packed (key,idx) u64).
// key = ordered-uint of (valid ? -score : +inf), tie-break by index -> exactly
// reproduces JAX stable argsort. One block (1024 thr) per class.
// ---------------------------------------------------------------------------
__global__ void sort_kernel(const float* __restrict__ cls_prob,
                            const float* __restrict__ pred,
                            float* __restrict__ sboxes,   // [20][N][4]
                            float* __restrict__ sscores)  // [20][N]
{
    const int j = blockIdx.x;      // 0..19
    const int c = j + 1;           // class column
    const int tid = threadIdx.x;   // 0..1023
    __shared__ unsigned long long keys[2048];

    for (int e = tid; e < 2048; e += 1024) {
        unsigned long long kv;
        if (e < N_ROIS) {
            float s = cls_prob[e * N_CLS + c];
            float fkey = (s > SCORE_THR) ? -s : __builtin_inff();
            unsigned u = __float_as_uint(fkey);
            u = (u & 0x80000000u) ? ~u : (u | 0x80000000u);  // order-preserving map
            kv = ((unsigned long long)u << 32) | (unsigned)e;
        } else {
            kv = 0xFFFFFFFFFFFFFFFFull;  // pad sorts last
        }
        keys[e] = kv;
    }
    __syncthreads();

    for (int k = 2; k <= 2048; k <<= 1) {
        for (int jj = k >> 1; jj > 0; jj >>= 1) {
            for (int e = tid; e < 2048; e += 1024) {
                int partner = e ^ jj;
                if (partner > e) {
                    bool asc = ((e & k) == 0);
                    unsigned long long a = keys[e], b = keys[partner];
                    bool swap = asc ? (a > b) : (a < b);
                    if (swap) { keys[e] = b; keys[partner] = a; }
                }
            }
            __syncthreads();
        }
    }

    for (int e = tid; e < N_ROIS; e += 1024) {
        unsigned idx = (unsigned)(keys[e] & 0xFFFFFFFFu);
        const float* b = pred + ((size_t)c * N_ROIS + idx) * 4;
        float* o = sboxes + ((size_t)j * N_ROIS + e) * 4;
        o[0] = b[0]; o[1] = b[1]; o[2] = b[2]; o[3] = b[3];
        sscores[(size_t)j * N_ROIS + e] = cls_prob[idx * N_CLS + c];
    }
}

// ---------------------------------------------------------------------------
// Kernel 3: sequential NMS scan (LDS-resident) + GT dedup + masked output.
// One block (256 thr) per class. Writes all 2000*5 outputs for its class.
// ---------------------------------------------------------------------------
__global__ void nms_kernel(const float* __restrict__ sboxes,
                           const float* __restrict__ sscores,
                           const float* __restrict__ gt_boxes,  // [20][5]
                           const int*   __restrict__ num_boxes, // scalar
                           float* __restrict__ out)             // [20][N][5]
{
    const int j = blockIdx.x;
    const int tid = threadIdx.x;
    const int bdim = blockDim.x;

    __shared__ float bx[N_ROIS][4];
    __shared__ float sc[N_ROIS];
    __shared__ int   keep[N_ROIS];
    __shared__ int   sup;

    for (int e = tid; e < N_ROIS; e += bdim) {
        const float* b = sboxes + ((size_t)j * N_ROIS + e) * 4;
        bx[e][0] = b[0]; bx[e][1] = b[1]; bx[e][2] = b[2]; bx[e][3] = b[3];
        sc[e] = sscores[(size_t)j * N_ROIS + e];
        keep[e] = 0;
    }
    __syncthreads();

    // sequential keep scan: keep[i] = valid[i] && !any(kept j<i with IoU>thr)
    for (int i = 0; i < N_ROIS; ++i) {
        if (tid == 0) sup = 0;
        __syncthreads();
        float ax1 = bx[i][0], ay1 = bx[i][1], ax2 = bx[i][2], ay2 = bx[i][3];
        int s = 0;
        for (int q = tid; q < i; q += bdim) {
            if (keep[q]) {
                float v = iou_f(ax1, ay1, ax2, ay2, bx[q][0], bx[q][1], bx[q][2], bx[q][3]);
                if (v > NMS_THR) s = 1;
            }
        }
        if (s) atomicOr(&sup, 1);
        __syncthreads();
        if (tid == 0) keep[i] = (sc[i] > SCORE_THR) && (sup == 0);
        __syncthreads();
    }

    // GT dedup + write (every output element written -> no stale poison)
    int nb = num_boxes[0];
    if (nb > N_GT) nb = N_GT;
    for (int i = tid; i < N_ROIS; i += bdim) {
        float ax1 = bx[i][0], ay1 = bx[i][1], ax2 = bx[i][2], ay2 = bx[i][3];
        bool hit = false;
        for (int m = 0; m < nb; ++m) {
            const float* g = gt_boxes + m * 5;
            if (iou_f(ax1, ay1, ax2, ay2, g[0], g[1], g[2], g[3]) > DEDUP_THR) hit = true;
        }
        bool fin = keep[i] && !hit;
        float* o = out + ((size_t)j * N_ROIS + i) * 5;
        o[0] = fin ? ax1 : 0.0f;
        o[1] = fin ? ay1 : 0.0f;
        o[2] = fin ? ax2 : 0.0f;
        o[3] = fin ? ay2 : 0.0f;
        o[4] = fin ? sc[i] : 0.0f;
    }
}

// ---------------------------------------------------------------------------
// Kernel 4: channel-mean norm losses via V_WMMA_F32_16X16X4_F32.
// mean over 512 channels == ones(16x4) x B(4ch x 16pos), accumulated over
// 128 K-steps. A==ones makes the result invariant to B's K-slot permutation.
// One block, 8 waves; waves stride over 119 position-tiles of 16.
// ---------------------------------------------------------------------------
__global__ void loss_kernel(const float* __restrict__ f,
                            const float* __restrict__ fo,
                            const float* __restrict__ fr,
                            float* __restrict__ out) // out[0],out[1]
{
    const int tid  = threadIdx.x;      // 256
    const int lane = tid & 31;
    const int wave = tid >> 5;         // 0..7
    const int n    = lane & 15;        // position within tile (column N)
    const int kh   = (lane >> 4) << 1; // lanes 0-15: k+0,k+1 ; 16-31: k+2,k+3
    const int ntiles = (FEAT_P + 15) / 16; // 119

    v2f ones; ones.x = 1.0f; ones.y = 1.0f;

    float s1 = 0.f, s2 = 0.f, s3 = 0.f, s4 = 0.f, s5 = 0.f;

    for (int t = wave; t < ntiles; t += 8) {
        int p  = t * 16 + n;
        int pc = (p < FEAT_P) ? p : (FEAT_P - 1);   // clamped, masked later
        v8f af = {}, ao = {}, ar = {};
        for (int k = 0; k < FEAT_C; k += 4) {
            int b0 = (k + kh) * FEAT_P + pc;
            int b1 = b0 + FEAT_P;
            v2f Bf, Bo, Br;
            Bf.x = f[b0];  Bf.y = f[b1];
            Bo.x = fo[b0]; Bo.y = fo[b1];
            Br.x = fr[b0]; Br.y = fr[b1];
            af = __builtin_amdgcn_wmma_f32_16x16x4_f32(false, ones, false, Bf,
                                                       (short)0, af, false, false);
            ao = __builtin_amdgcn_wmma_f32_16x16x4_f32(false, ones, false, Bo,
                                                       (short)0, ao, false, false);
            ar = __builtin_amdgcn_wmma_f32_16x16x4_f32(false, ones, false, Br,
                                                       (short)0, ar, false, false);
        }
        // Every lane's element 0 holds the full column (channel) sum for its n.
        float mf = af[0] * (1.0f / FEAT_C);
        float mo = ao[0] * (1.0f / FEAT_C);
        float mr = ar[0] * (1.0f / FEAT_C);
        float valid = ((p < FEAT_P) && (lane < 16)) ? 1.0f : 0.0f; // count once
        s1 += valid * mf * mf;
        s2 += valid * mo * mo;
        float dd = mf - mo; s3 += valid * dd * dd;
        s4 += valid * mr * mr;
        float ee = mo + mr; s5 += valid * ee * ee;
    }

    __shared__ float red[256][5];
    red[tid][0] = s1; red[tid][1] = s2; red[tid][2] = s3;
    red[tid][3] = s4; red[tid][4] = s5;
    __syncthreads();
    for (int off = 128; off > 0; off >>= 1) {
        if (tid < off) {
            for (int cidx = 0; cidx < 5; ++cidx)
                red[tid][cidx] += red[tid + off][cidx];
        }
        __syncthreads();
    }
    if (tid == 0) {
        float n_f   = sqrtf(red[0][0]);
        float n_fo  = sqrtf(red[0][1]);
        float n_dif = sqrtf(red[0][2]);
        float n_fr  = sqrtf(red[0][3]);
        float n_sum = sqrtf(red[0][4]);
        float distil = fabsf(n_f - n_fo);
        float res_loss     = fabsf(n_dif - n_fr);
        float res_loss_inc = fabsf(n_sum - n_f);
        out[0] = distil;
        out[1] = res_loss_inc + res_loss;
    }
}

// ---------------------------------------------------------------------------
extern "C" void kernel_launch(void* const* d_in, const int* in_sizes, int n_in,
                              void* d_out, int out_size, void* d_ws, size_t ws_size,
                              hipStream_t stream) {
    const float* rois      = (const float*)d_in[0];
    const float* cls_prob  = (const float*)d_in[1];
    const float* bbox_pred = (const float*)d_in[2];
    const float* im_info   = (const float*)d_in[3];
    const float* gt_boxes  = (const float*)d_in[4];
    const int*   num_boxes = (const int*)  d_in[5];
    const float* feat      = (const float*)d_in[6];
    const float* feat_org  = (const float*)d_in[7];
    const float* feat_res  = (const float*)d_in[8];
    float* out = (float*)d_out;

    // workspace layout (floats)
    float* ws      = (float*)d_ws;
    float* pred    = ws;                         // 21*2000*4 = 168000
    float* sboxes  = ws + 168000;                // 20*2000*4 = 160000
    float* sscores = ws + 328000;                // 20*2000   =  40000

    // 1) decode all N*C boxes
    {
        int total = N_ROIS * N_CLS;
        decode_kernel<<<(total + 255) / 256, 256, 0, stream>>>(rois, bbox_pred, im_info, pred);
    }
    // 2) per-class stable sort
    sort_kernel<<<N_JCLS, 1024, 0, stream>>>(cls_prob, pred, sboxes, sscores);
    // 3) per-class NMS + dedup + dets
    nms_kernel<<<N_JCLS, 256, 0, stream>>>(sboxes, sscores, gt_boxes, num_boxes, out);
    // 4) WMMA channel-mean losses -> last two outputs
    loss_kernel<<<1, 256, 0, stream>>>(feat, feat_org, feat_res,
                                       out + (size_t)N_JCLS * N_ROIS * 5);
}